// TaskAttention_15247133900833
// MI455X (gfx1250) — compile-verified
//
#include <hip/hip_runtime.h>
#include <math.h>

// ---------------------------------------------------------------------------
// TaskAttention on MI455X (gfx1250).
// Sparsity-aware rewrite: only the score GEMM (4096x48x768 per batch) is real
// compute; it runs on V_WMMA_F32_16X16X4_F32 (exact fp32, no precision loss
// for the top-k selection). Everything else is tiny or a 25MB stream.
// ---------------------------------------------------------------------------

#define NHEADS 12
#define NTASKS 4
#define TOPK   2
#define BB     2
#define NN     4100
#define CC     768
#define MM     4096
#define DD     64
#define TH     48            // NTASKS * NHEADS
#define SCALE  0.125f        // d^-0.5 = 1/8

typedef float v2f __attribute__((ext_vector_type(2)));
typedef float v8f __attribute__((ext_vector_type(8)));

// --------------------------- kernel 0: zero output -------------------------
__global__ void k_zero(float4* out, int n4) {
    int i = blockIdx.x * blockDim.x + threadIdx.x;
    if (i < n4) out[i] = make_float4(0.f, 0.f, 0.f, 0.f);
}

// ------------------- kernel 1a: q[b,t,o] = x[b,t,:] . qs_w[t,o,:] ----------
__global__ void k_qproj(const float* __restrict__ x,
                        const float* __restrict__ qs_w,
                        float* __restrict__ q) {
    int bt = blockIdx.x;               // 0..7  (b*4+t)
    int t  = bt & 3;
    int b  = bt >> 2;
    __shared__ float xs[CC];
    const float* xr = x + ((size_t)b * NN + t) * CC;
    for (int c = threadIdx.x; c < CC; c += 256) xs[c] = xr[c];
    __syncthreads();
    const float* W = qs_w + (size_t)t * CC * CC;
    for (int o = threadIdx.x; o < CC; o += 256) {
        const float* wr = W + (size_t)o * CC;
        float acc = 0.f;
        for (int c = 0; c < CC; ++c) acc += xs[c] * wr[c];
        q[bt * CC + o] = acc;
    }
}

// ---- kernel 1b: qk[b, t*H+h, c] = scale * sum_dd q[b,t,h*64+dd]*kv_w[h*64+dd, c]
__global__ void k_qkfold(const float* __restrict__ q,
                         const float* __restrict__ kv_w,
                         float* __restrict__ qk) {
    int e  = blockIdx.x;               // 0..95 (b,t,h)
    int h  = e % NHEADS;
    int bt = e / NHEADS;
    int t  = bt & 3, b = bt >> 2;
    __shared__ float qs[DD];
    if (threadIdx.x < DD) qs[threadIdx.x] = q[bt * CC + h * DD + threadIdx.x];
    __syncthreads();
    float* dst = qk + ((size_t)b * TH + t * NHEADS + h) * CC;
    for (int c = threadIdx.x; c < CC; c += 256) {
        const float* kr = kv_w + (size_t)(h * DD) * CC + c;  // k-half rows
        float acc = 0.f;
        for (int dd = 0; dd < DD; ++dd) acc += qs[dd] * kr[(size_t)dd * CC];
        dst[c] = acc * SCALE;
    }
}

// -------- kernel 2: S[b,th,m] = feature[b,m,:] . qk[b,th,:]  (WMMA fp32) ---
// One wave per 16-row m-tile; 3 n-tiles (48 th columns); K loop = 768/4.
__global__ void k_scores(const float* __restrict__ x,
                         const float* __restrict__ qk,
                         float* __restrict__ S) {
    int wid  = blockIdx.x;             // 0..511
    int b    = wid >> 8;
    int m0   = (wid & 255) * 16;
    int lane = threadIdx.x;            // wave32
    int r    = lane & 15;
    int half = lane >> 4;

    // A frag: lane holds row m0+r; K pairs {0,1} (lanes 0-15) / {2,3} (16-31)
    const float* feat = x + ((size_t)b * NN + NTASKS) * CC;
    const float* arow = feat + (size_t)(m0 + r) * CC + 2 * half;
    // B frag: lane holds column n = th; same K striping (ISA 7.12.2 mirror)
    const float* qb = qk + (size_t)b * TH * CC;
    const float* br0 = qb + (size_t)(r)      * CC + 2 * half;
    const float* br1 = qb + (size_t)(16 + r) * CC + 2 * half;
    const float* br2 = qb + (size_t)(32 + r) * CC + 2 * half;

    v8f acc0 = {}, acc1 = {}, acc2 = {};
    for (int k0 = 0; k0 < CC; k0 += 4) {
        v2f a  = *(const v2f*)(arow + k0);
        v2f b0 = *(const v2f*)(br0 + k0);
        v2f b1 = *(const v2f*)(br1 + k0);
        v2f b2 = *(const v2f*)(br2 + k0);
        acc0 = __builtin_amdgcn_wmma_f32_16x16x4_f32(false, a, false, b0, (short)0, acc0, false, false);
        acc1 = __builtin_amdgcn_wmma_f32_16x16x4_f32(false, a, false, b1, (short)0, acc1, false, false);
        acc2 = __builtin_amdgcn_wmma_f32_16x16x4_f32(false, a, false, b2, (short)0, acc2, false, false);
    }
    // D layout: lane L -> column n=(L&15); VGPR i -> row i + 8*(L>=16)
    float* Sb = S + (size_t)b * TH * MM;
    int mbase = m0 + 8 * half;
    for (int i = 0; i < 8; ++i) {
        Sb[(size_t)(r)      * MM + mbase + i] = acc0[i];
        Sb[(size_t)(16 + r) * MM + mbase + i] = acc1[i];
        Sb[(size_t)(32 + r) * MM + mbase + i] = acc2[i];
    }
}

// ---- kernel 3: per (b,th): top-2 over m, softmax, on-demand v, attn_token -
__global__ void k_topk(const float* __restrict__ S,
                       const float* __restrict__ x,
                       const float* __restrict__ kv_w,
                       float* __restrict__ attn_token,
                       int* __restrict__ m_idx,
                       float* __restrict__ wgt) {
    int e  = blockIdx.x;               // 0..95
    int b  = e / TH;
    int th = e % TH;
    int t  = th / NHEADS;
    int h  = th % NHEADS;
    int tid = threadIdx.x;

    const float* Srow = S + ((size_t)b * TH + th) * MM;
    float v1 = -INFINITY, v2 = -INFINITY;
    int   i1 = 0x7fffffff, i2 = 0x7fffffff;
    for (int m = tid; m < MM; m += 256) {
        float s = Srow[m];
        if (s > v1 || (s == v1 && m < i1)) { v2 = v1; i2 = i1; v1 = s; i1 = m; }
        else if (s > v2 || (s == v2 && m < i2)) { v2 = s; i2 = m; }
    }
    __shared__ float sv1[256], sv2[256];
    __shared__ int   si1[256], si2[256];
    sv1[tid] = v1; sv2[tid] = v2; si1[tid] = i1; si2[tid] = i2;
    __syncthreads();
    for (int stride = 128; stride > 0; stride >>= 1) {
        if (tid < stride) {
            float a1 = sv1[tid], a2 = sv2[tid];
            float b1 = sv1[tid + stride], b2 = sv2[tid + stride];
            int  ai1 = si1[tid], ai2 = si2[tid];
            int  bi1 = si1[tid + stride], bi2 = si2[tid + stride];
            float n1, n2; int ni1, ni2;
            bool bwin = (b1 > a1) || (b1 == a1 && bi1 < ai1);
            if (bwin) {
                n1 = b1; ni1 = bi1;
                bool awin = (a1 > b2) || (a1 == b2 && ai1 < bi2);
                if (awin) { n2 = a1; ni2 = ai1; } else { n2 = b2; ni2 = bi2; }
            } else {
                n1 = a1; ni1 = ai1;
                bool bwin2 = (b1 > a2) || (b1 == a2 && bi1 < ai2);
                if (bwin2) { n2 = b1; ni2 = bi1; } else { n2 = a2; ni2 = ai2; }
            }
            sv1[tid] = n1; sv2[tid] = n2; si1[tid] = ni1; si2[tid] = ni2;
        }
        __syncthreads();
    }
    __shared__ float w_sh[2];
    __shared__ int   mi_sh[2];
    if (tid == 0) {
        float ex = expf(sv2[0] - sv1[0]);     // v2 <= v1: stable softmax
        float w1 = 1.f / (1.f + ex);
        w_sh[0] = w1; w_sh[1] = ex * w1;
        mi_sh[0] = si1[0]; mi_sh[1] = si2[0];
        int base = e * TOPK;
        m_idx[base] = si1[0]; m_idx[base + 1] = si2[0];
        wgt[base] = w_sh[0];  wgt[base + 1] = w_sh[1];
    }
    __syncthreads();

    // stage the two selected feature rows, then compute v only at those rows
    __shared__ float frow[TOPK][CC];
    const float* feat = x + ((size_t)b * NN + NTASKS) * CC;
    for (int k = 0; k < TOPK; ++k)
        for (int c = tid; c < CC; c += 256)
            frow[k][c] = feat[(size_t)mi_sh[k] * CC + c];
    __syncthreads();
    __shared__ float vpart[TOPK][DD];
    if (tid < TOPK * DD) {
        int k = tid >> 6, dd = tid & 63;
        const float* kr = kv_w + (size_t)(CC + h * DD + dd) * CC;  // v-half rows
        float acc = 0.f;
        for (int c = 0; c < CC; ++c) acc += frow[k][c] * kr[c];
        vpart[k][dd] = acc;
    }
    __syncthreads();
    if (tid < DD)
        attn_token[((size_t)b * NTASKS + t) * CC + h * DD + tid] =
            w_sh[0] * vpart[0][tid] + w_sh[1] * vpart[1][tid];
}

// ------- kernel 4: token_output[b,t,o] = attn_token . experts_w[t,o,:] -----
__global__ void k_token_out(const float* __restrict__ attn_token,
                            const float* __restrict__ experts_w,
                            float* __restrict__ out) {
    int bt = blockIdx.x;               // 0..7
    int t  = bt & 3, b = bt >> 2;
    __shared__ float at[CC];
    for (int c = threadIdx.x; c < CC; c += 256) at[c] = attn_token[bt * CC + c];
    __syncthreads();
    const float* W = experts_w + (size_t)t * CC * CC;
    float* dst = out + ((size_t)b * NN + t) * CC;
    for (int o = threadIdx.x; o < CC; o += 256) {
        const float* wr = W + (size_t)o * CC;
        float acc = 0.f;
        for (int c = 0; c < CC; ++c) acc += at[c] * wr[c];
        dst[o] = acc;
    }
}

// --- kernel 5: sparse feature_output scatter; each thread owns one column o,
// loops the 96 (th,k) entries serially -> race-free, bit-deterministic -------
__global__ void k_feat_out(const float* __restrict__ x,
                           const float* __restrict__ experts_w,
                           const int* __restrict__ m_idx,
                           const float* __restrict__ wgt,
                           float* __restrict__ out) {
    int blk = blockIdx.x;              // 0..5  (b * 3 + chunk)
    int b = blk / 3;
    int o = (blk % 3) * 256 + threadIdx.x;
    for (int e = 0; e < TH * TOPK; ++e) {
        int idx = b * TH * TOPK + e;
        int th  = e >> 1;
        int t   = th / NHEADS, h = th % NHEADS;
        int m   = m_idx[idx];
        float w = wgt[idx];
        const float* fr = x + ((size_t)b * NN + NTASKS + m) * CC + h * DD;
        const float* wr = experts_w + ((size_t)t * CC + o) * CC + h * DD;
        float acc = 0.f;
        for (int dd = 0; dd < DD; ++dd) acc += fr[dd] * wr[dd];
        float* dp = out + ((size_t)b * NN + NTASKS + m) * CC + o;
        *dp += w * acc;                // thread-private column: no races
    }
}

// ---------------------------------------------------------------------------
extern "C" void kernel_launch(void* const* d_in, const int* in_sizes, int n_in,
                              void* d_out, int out_size, void* d_ws, size_t ws_size,
                              hipStream_t stream) {
    const float* x         = (const float*)d_in[0];
    const float* qs_w      = (const float*)d_in[1];
    const float* kv_w      = (const float*)d_in[2];
    const float* experts_w = (const float*)d_in[3];
    float* out = (float*)d_out;

    // workspace layout (floats), ~1.9 MB total
    float* q          = (float*)d_ws;                       // 8*768
    float* qk         = q + BB * NTASKS * CC;               // 2*48*768
    float* S          = qk + BB * TH * CC;                  // 2*48*4096
    float* attn_token = S + (size_t)BB * TH * MM;           // 8*768
    int*   m_idx      = (int*)(attn_token + BB * NTASKS * CC); // 192
    float* wgt        = (float*)(m_idx + BB * TH * TOPK);      // 192

    int n4 = out_size / 4;
    k_zero     <<<(n4 + 255) / 256, 256, 0, stream>>>((float4*)out, n4);
    k_qproj    <<<BB * NTASKS,            256, 0, stream>>>(x, qs_w, q);
    k_qkfold   <<<BB * NTASKS * NHEADS,   256, 0, stream>>>(q, kv_w, qk);
    k_scores   <<<BB * 256,                32, 0, stream>>>(x, qk, S);
    k_topk     <<<BB * TH,                256, 0, stream>>>(S, x, kv_w, attn_token, m_idx, wgt);
    k_token_out<<<BB * NTASKS,            256, 0, stream>>>(attn_token, experts_w, out);
    k_feat_out <<<BB * 3,                 256, 0, stream>>>(x, experts_w, m_idx, wgt, out);
}